// BSM_78726750536308
// MI455X (gfx1250) — compile-verified
//
#include <hip/hip_runtime.h>
#include <cstdint>

typedef float v2f __attribute__((ext_vector_type(2)));
typedef float v8f __attribute__((ext_vector_type(8)));

#define B_N 16
#define L_N 32768
#define NCHUNK 256
#define CHUNK 128   /* = L_N / NCHUNK */
#define EPSF 1e-5f
#define WPB 4       /* waves per block in scan kernels */
#define WSLOT 768   /* per-wave LDS slice (floats) */

__device__ __forceinline__ float sp_softplus(float x) {
    // branchless, stable: max(x,0) + log(1 + exp(-|x|)); fast transcendentals
    return fmaxf(x, 0.f) + __logf(1.f + __expf(-fabsf(x)));
}
__device__ __forceinline__ float silu_f(float x) {
    return x * (1.f / (1.f + __expf(-x)));
}

// ---------------------------------------------------------------------------
// K1: residual rmsnorm -> in_proj -> causal depthwise conv (fwd + reversed)
//     -> SiLU. Produces u_f (fwd layout), u_b (reversed layout), g = silu(z).
// ---------------------------------------------------------------------------
__global__ void prep_kernel(const float* __restrict__ hs,
                            const float* __restrict__ norm_w,
                            const float* __restrict__ ipw,
                            const float* __restrict__ cwf, const float* __restrict__ cbf,
                            const float* __restrict__ cwb, const float* __restrict__ cbb,
                            float* __restrict__ uf, float* __restrict__ ub,
                            float* __restrict__ g)
{
    int idx = blockIdx.x * blockDim.x + threadIdx.x;
    if (idx >= B_N * L_N) return;
    int b = idx / L_N, l = idx - b * L_N;
    const float nw0 = norm_w[0], nw1 = norm_w[1];
    float X[7][4];
    float Z[4] = {0.f, 0.f, 0.f, 0.f};
    #pragma unroll
    for (int o = 0; o < 7; ++o) {
        int ll = l - 3 + o;
        if (ll < 0 || ll >= L_N) {
            X[o][0] = X[o][1] = X[o][2] = X[o][3] = 0.f;
            continue;
        }
        float r0 = hs[(size_t)(b * 2 + 0) * L_N + ll];
        float r1 = hs[(size_t)(b * 2 + 1) * L_N + ll];
        float s = rsqrtf(0.5f * (r0 * r0 + r1 * r1) + EPSF);
        float n0 = r0 * s * nw0, n1 = r1 * s * nw1;
        #pragma unroll
        for (int j = 0; j < 4; ++j)
            X[o][j] = ipw[j * 2 + 0] * n0 + ipw[j * 2 + 1] * n1;
        if (o == 3) {
            #pragma unroll
            for (int j = 0; j < 4; ++j)
                Z[j] = ipw[(4 + j) * 2 + 0] * n0 + ipw[(4 + j) * 2 + 1] * n1;
        }
    }
    size_t obase_f = ((size_t)b * L_N + l) * 4;
    int r = L_N - 1 - l;
    size_t obase_b = ((size_t)b * L_N + r) * 4;
    #pragma unroll
    for (int d = 0; d < 4; ++d) {
        float accf = cbf[d], accb = cbb[d];
        #pragma unroll
        for (int k = 0; k < 4; ++k) {
            accf += cwf[d * 4 + k] * X[k][d];       // x[l-3+k]
            accb += cwb[d * 4 + k] * X[6 - k][d];   // x_rev[r-3+k] = x[l+3-k]
        }
        uf[obase_f + d] = silu_f(accf);
        ub[obase_b + d] = silu_f(accb);
        g[obase_f + d]  = silu_f(Z[d]);
    }
}

// ---------------------------------------------------------------------------
// Scan kernel. One wave per (b, dir, chunk). Lane = (n = lane&15, half =
// lane>>4); lane owns state h[d0=half][n] and h[d1=half+2][n].
// Per 16-step tile: x_proj B/C tiles via V_WMMA_F32_16X16X4_F32, results
// staged through a per-wave LDS slice (conflict-free stride-17 rows) so the
// v8f tiles die immediately (no spills) and the step loop only does cheap
// ds_load broadcasts. softplus/dt*u precomputed once per tile (2/lane).
// PHASE3=false: compute P=prod(dA) and local end state.
// PHASE3=true : rerun from h_start, emit gated y.
// ---------------------------------------------------------------------------
template <bool PHASE3>
__global__ void scan_kernel(const float* __restrict__ uf, const float* __restrict__ ub,
                            const float* __restrict__ g,
                            const float* __restrict__ xpw_f, const float* __restrict__ dtw_f,
                            const float* __restrict__ dtb_f, const float* __restrict__ Alog_f,
                            const float* __restrict__ Dv_f,
                            const float* __restrict__ xpw_b, const float* __restrict__ dtw_b,
                            const float* __restrict__ dtb_b, const float* __restrict__ Alog_b,
                            const float* __restrict__ Dv_b,
                            float* __restrict__ Pbuf, float* __restrict__ hendbuf,
                            const float* __restrict__ hstartbuf,
                            float* __restrict__ yf, float* __restrict__ yb)
{
    __shared__ float smem[WPB * WSLOT];
    const int lane = threadIdx.x & 31;
    const int wib  = threadIdx.x >> 5;
    const int gw = blockIdx.x * WPB + wib;
    const int chunk = gw % NCHUNK;
    const int dir = (gw / NCHUNK) & 1;
    const int b = gw / (NCHUNK * 2);
    const int n = lane & 15, half = lane >> 4;
    const int d0 = half, d1 = half + 2;

    // per-wave LDS slice
    float* u_sm  = smem + wib * WSLOT;      // 64:  u[t][d]
    float* dt_sm = u_sm + 64;               // 64:  dt[t][d]
    float* du_sm = u_sm + 128;              // 64:  dt[t][d]*u[t][d]
    float* B_sm  = u_sm + 192;              // 272: B[t][n] at [n*17+t]
    float* C_sm  = u_sm + 464;              // 272: C[t][n] at [n*17+t]

    const float* xpw  = dir ? xpw_b  : xpw_f;
    const float* dtw  = dir ? dtw_b  : dtw_f;
    const float* dtb  = dir ? dtb_b  : dtb_f;
    const float* Alog = dir ? Alog_b : Alog_f;
    const float* Dv   = dir ? Dv_b   : Dv_f;
    const float* u    = dir ? ub : uf;
    float* ybuf       = dir ? yb : yf;

    // B operand of WMMA: lane (n, half) holds W[K=2*half+r][n] for r=0,1.
    // x_proj_w rows: 0 = dt, 1..16 = B, 17..32 = C; cols = K (d_inner).
    v2f Bw  = *(const v2f*)(xpw + (1 + n) * 4 + 2 * half);
    v2f Cw  = *(const v2f*)(xpw + (17 + n) * 4 + 2 * half);
    v2f Wdt = *(const v2f*)(xpw + 2 * half);
    const float dtw0 = dtw[d0], dtw1 = dtw[d1];
    const float dtb0 = dtb[d0], dtb1 = dtb[d1];
    const float A0 = -__expf(Alog[d0 * 16 + n]);
    const float A1 = -__expf(Alog[d1 * 16 + n]);
    const float Dd0 = Dv[d0], Dd1 = Dv[d1];

    const int sid = b * 2 + dir;
    const size_t coff = ((size_t)(sid * NCHUNK + chunk)) * 64;
    float h0, h1, P0 = 1.f, P1 = 1.f;
    if (PHASE3) {
        h0 = hstartbuf[coff + d0 * 16 + n];
        h1 = hstartbuf[coff + d1 * 16 + n];
    } else {
        h0 = 0.f; h1 = 0.f;
    }

    const size_t ubase = ((size_t)b * L_N + (size_t)chunk * CHUNK);

    for (int tt = 0; tt < CHUNK / 16; ++tt) {
        // A operand: 16x4 u tile. Lane (m=n, half) holds u[m][2*half+r], r=0,1.
        v2f a = *(const v2f*)(u + (ubase + tt * 16 + n) * 4 + 2 * half);

        // stage u tile to LDS: u_sm[t*4+d]
        u_sm[n * 4 + 2 * half + 0] = a.x;
        u_sm[n * 4 + 2 * half + 1] = a.y;

        // rank-1 dt projection: dtraw[m] replicated on lanes m, m+16
        float pd = a.x * Wdt.x + a.y * Wdt.y;
        pd += __shfl_xor(pd, 16, 32);

        // per-tile softplus precompute: this lane covers (t=n, d0) and (t=n, d1)
        float sp0 = sp_softplus(fmaf(pd, dtw0, dtb0));
        float sp1 = sp_softplus(fmaf(pd, dtw1, dtb1));
        dt_sm[n * 4 + d0] = sp0;
        dt_sm[n * 4 + d1] = sp1;
        du_sm[n * 4 + d0] = sp0 * u_sm[n * 4 + d0];
        du_sm[n * 4 + d1] = sp1 * u_sm[n * 4 + d1];

        v8f zero8 = {0.f, 0.f, 0.f, 0.f, 0.f, 0.f, 0.f, 0.f};
        v8f Bt = __builtin_amdgcn_wmma_f32_16x16x4_f32(
            false, a, false, Bw, (short)0, zero8, false, false);
        // D-layout: lane (n, half) holds rows t = v + 8*half, v=0..7, column n
        #pragma unroll
        for (int v = 0; v < 8; ++v)
            B_sm[n * 17 + half * 8 + v] = Bt[v];
        if (PHASE3) {
            v8f Ct = __builtin_amdgcn_wmma_f32_16x16x4_f32(
                false, a, false, Cw, (short)0, zero8, false, false);
            #pragma unroll
            for (int v = 0; v < 8; ++v)
                C_sm[n * 17 + half * 8 + v] = Ct[v];
        }

        #pragma unroll
        for (int t = 0; t < 16; ++t) {
            float Bv  = B_sm[n * 17 + t];       // broadcast to both halves
            float dt0 = dt_sm[t * 4 + d0];      // LDS broadcast within half
            float dt1 = dt_sm[t * 4 + d1];
            float du0 = du_sm[t * 4 + d0];
            float du1 = du_sm[t * 4 + d1];
            float dA0 = __expf(dt0 * A0);
            float dA1 = __expf(dt1 * A1);
            h0 = fmaf(dA0, h0, du0 * Bv);
            h1 = fmaf(dA1, h1, du1 * Bv);
            if (!PHASE3) { P0 *= dA0; P1 *= dA1; }
            if (PHASE3) {
                float Cv = C_sm[n * 17 + t];
                float p0 = h0 * Cv, p1 = h1 * Cv;
                #pragma unroll
                for (int m = 1; m <= 8; m <<= 1) {
                    p0 += __shfl_xor(p0, m, 32);
                    p1 += __shfl_xor(p1, m, 32);
                }
                if (n == t) {
                    int pos = chunk * CHUNK + tt * 16 + t;      // scan coordinate
                    int lfwd = dir ? (L_N - 1 - pos) : pos;     // forward coordinate
                    size_t gb = ((size_t)b * L_N + lfwd) * 4;
                    float y0 = fmaf(u_sm[t * 4 + d0], Dd0, p0) * g[gb + d0];
                    float y1 = fmaf(u_sm[t * 4 + d1], Dd1, p1) * g[gb + d1];
                    size_t yo = ((size_t)b * L_N + pos) * 4;
                    ybuf[yo + d0] = y0;
                    ybuf[yo + d1] = y1;
                }
            }
        }
    }
    if (!PHASE3) {
        Pbuf[coff + d0 * 16 + n]    = P0;
        Pbuf[coff + d1 * 16 + n]    = P1;
        hendbuf[coff + d0 * 16 + n] = h0;
        hendbuf[coff + d1 * 16 + n] = h1;
    }
}

// ---------------------------------------------------------------------------
// K3: sequential combine of chunk summaries. 32 blocks x 32 lanes.
// ---------------------------------------------------------------------------
__global__ void combine_kernel(const float* __restrict__ Pbuf,
                               const float* __restrict__ hendbuf,
                               float* __restrict__ hstartbuf)
{
    int sid = blockIdx.x;
    int lane = threadIdx.x;
    int n = lane & 15, half = lane >> 4;
    int e0 = half * 16 + n, e1 = (half + 2) * 16 + n;
    float h0 = 0.f, h1 = 0.f;
    for (int c = 0; c < NCHUNK; ++c) {
        size_t off = ((size_t)(sid * NCHUNK + c)) * 64;
        hstartbuf[off + e0] = h0;
        hstartbuf[off + e1] = h1;
        h0 = Pbuf[off + e0] * h0 + hendbuf[off + e0];
        h1 = Pbuf[off + e1] * h1 + hendbuf[off + e1];
    }
}

// ---------------------------------------------------------------------------
// K5: y_f + reversed y_b -> out_proj -> +residual -> rmsnorm -> output (B,T,L)
// ---------------------------------------------------------------------------
__global__ void final_kernel(const float* __restrict__ hs,
                             const float* __restrict__ yf, const float* __restrict__ yb,
                             const float* __restrict__ opw, const float* __restrict__ nfw,
                             float* __restrict__ out)
{
    int idx = blockIdx.x * blockDim.x + threadIdx.x;
    if (idx >= B_N * L_N) return;
    int b = idx / L_N, l = idx - b * L_N;
    int r = L_N - 1 - l;
    size_t fo = ((size_t)b * L_N + l) * 4;
    size_t bo = ((size_t)b * L_N + r) * 4;
    float y0 = yf[fo + 0] + yb[bo + 0];
    float y1 = yf[fo + 1] + yb[bo + 1];
    float y2 = yf[fo + 2] + yb[bo + 2];
    float y3 = yf[fo + 3] + yb[bo + 3];
    float o0 = opw[0] * y0 + opw[1] * y1 + opw[2] * y2 + opw[3] * y3
             + hs[(size_t)(b * 2 + 0) * L_N + l];
    float o1 = opw[4] * y0 + opw[5] * y1 + opw[6] * y2 + opw[7] * y3
             + hs[(size_t)(b * 2 + 1) * L_N + l];
    float s = rsqrtf(0.5f * (o0 * o0 + o1 * o1) + EPSF);
    out[(size_t)(b * 2 + 0) * L_N + l] = o0 * s * nfw[0];
    out[(size_t)(b * 2 + 1) * L_N + l] = o1 * s * nfw[1];
}

extern "C" void kernel_launch(void* const* d_in, const int* in_sizes, int n_in,
                              void* d_out, int out_size, void* d_ws, size_t ws_size,
                              hipStream_t stream)
{
    const float* hs    = (const float*)d_in[0];
    const float* norm_w = (const float*)d_in[1];
    const float* ipw   = (const float*)d_in[2];
    const float* cwf   = (const float*)d_in[3];
    const float* cbf   = (const float*)d_in[4];
    const float* xpwf  = (const float*)d_in[5];
    const float* dtwf  = (const float*)d_in[6];
    const float* dtbf  = (const float*)d_in[7];
    const float* Alogf = (const float*)d_in[8];
    const float* Df    = (const float*)d_in[9];
    const float* cwb   = (const float*)d_in[10];
    const float* cbb   = (const float*)d_in[11];
    const float* xpwb  = (const float*)d_in[12];
    const float* dtwb  = (const float*)d_in[13];
    const float* dtbb  = (const float*)d_in[14];
    const float* Alogb = (const float*)d_in[15];
    const float* Db    = (const float*)d_in[16];
    const float* opw   = (const float*)d_in[17];
    const float* nfw   = (const float*)d_in[18];
    float* out = (float*)d_out;

    float* ws = (float*)d_ws;
    const size_t SZ_U = (size_t)B_N * L_N * 4;            // 2M floats each
    const size_t SZ_C = (size_t)B_N * 2 * NCHUNK * 64;    // chunk summaries
    float* uf  = ws;
    float* ub  = uf + SZ_U;
    float* g   = ub + SZ_U;
    float* yf  = g  + SZ_U;
    float* yb  = yf + SZ_U;
    float* P   = yb + SZ_U;
    float* he  = P  + SZ_C;
    float* hst = he + SZ_C;

    const int total = B_N * L_N;
    prep_kernel<<<(total + 255) / 256, 256, 0, stream>>>(
        hs, norm_w, ipw, cwf, cbf, cwb, cbb, uf, ub, g);

    const int nwaves = B_N * 2 * NCHUNK;   // 8192 waves, WPB waves per block
    scan_kernel<false><<<nwaves / WPB, WPB * 32, 0, stream>>>(
        uf, ub, g, xpwf, dtwf, dtbf, Alogf, Df,
        xpwb, dtwb, dtbb, Alogb, Db, P, he, hst, yf, yb);

    combine_kernel<<<B_N * 2, 32, 0, stream>>>(P, he, hst);

    scan_kernel<true><<<nwaves / WPB, WPB * 32, 0, stream>>>(
        uf, ub, g, xpwf, dtwf, dtbf, Alogf, Df,
        xpwb, dtwb, dtbb, Alogb, Db, P, he, hst, yf, yb);

    final_kernel<<<(total + 255) / 256, 256, 0, stream>>>(hs, yf, yb, opw, nfw, out);
}